// GCNLayer_88527865905439
// MI455X (gfx1250) — compile-verified
//
#include <hip/hip_runtime.h>

typedef __attribute__((ext_vector_type(2))) float v2f;
typedef __attribute__((ext_vector_type(8))) float v8f;

#define N_NODES 100000
#define N_EDGES 1000000
#define D 64
#define ROW_TILES (N_NODES / 16) // 6250, N_NODES % 16 == 0

// ---------------- zero init (d_out poisoned by harness; deg needs zeros) ----
__global__ void zero_kernel(float* __restrict__ out, float* __restrict__ deg) {
    const size_t nOut = (size_t)N_NODES * D;
    const size_t total = nOut + N_NODES;
    for (size_t i = (size_t)blockIdx.x * blockDim.x + threadIdx.x; i < total;
         i += (size_t)gridDim.x * blockDim.x) {
        if (i < nOut) out[i] = 0.0f;
        else          deg[i - nOut] = 0.0f;
    }
}

// ---------------- degree: deg[dst] += 1 -------------------------------------
__global__ void degree_kernel(const int* __restrict__ dst, float* __restrict__ deg) {
    int e = blockIdx.x * blockDim.x + threadIdx.x;
    if (e < N_EDGES) unsafeAtomicAdd(&deg[dst[e]], 1.0f);
}

// ---------------- dinv = rsqrt(deg + 2) in place ----------------------------
__global__ void dinv_kernel(float* __restrict__ deg) {
    int i = blockIdx.x * blockDim.x + threadIdx.x;
    if (i < N_NODES) deg[i] = __frsqrt_rn(deg[i] + 2.0f);
}

// ---------------- xw = x @ W^T via V_WMMA_F32_16X16X4_F32 -------------------
// Per wave: one 16-row tile of x, full 64 output cols (4 WMMA accumulators).
// A (16x4 f32) layout: lanes 0-15 hold M=lane K={0,1}; lanes 16-31 hold K={2,3}.
// B (4x16 f32) layout mirrors A with N in lanes. B[k][n] = W[n][k] (row-major W).
// C/D (16x16 f32): VGPR j = row j (lanes 0-15) / row j+8 (lanes 16-31), N=lane%16.
__global__ void __launch_bounds__(256) gemm_wmma_kernel(const float* __restrict__ x,
                                                        const float* __restrict__ W,
                                                        float* __restrict__ xw) {
    const int wave = threadIdx.x >> 5;
    const int tile = blockIdx.x * 8 + wave;
    if (tile >= ROW_TILES) return;                 // whole-wave exit, EXEC stays all-1s
    const int lane = threadIdx.x & 31;
    const int half = lane >> 4;                    // 0: K=0,1   1: K=2,3
    const int l16  = lane & 15;

    const float* xr = x + (size_t)(tile * 16 + l16) * D + 2 * half;
    const float* w0 = W + (size_t)(l16 +  0) * D + 2 * half;
    const float* w1 = W + (size_t)(l16 + 16) * D + 2 * half;
    const float* w2 = W + (size_t)(l16 + 32) * D + 2 * half;
    const float* w3 = W + (size_t)(l16 + 48) * D + 2 * half;

    v8f c0 = {}, c1 = {}, c2 = {}, c3 = {};
#pragma unroll
    for (int k = 0; k < D; k += 4) {
        v2f a  = *(const v2f*)(xr + k);
        v2f b0 = *(const v2f*)(w0 + k);
        v2f b1 = *(const v2f*)(w1 + k);
        v2f b2 = *(const v2f*)(w2 + k);
        v2f b3 = *(const v2f*)(w3 + k);
        c0 = __builtin_amdgcn_wmma_f32_16x16x4_f32(false, a, false, b0, (short)0, c0, false, false);
        c1 = __builtin_amdgcn_wmma_f32_16x16x4_f32(false, a, false, b1, (short)0, c1, false, false);
        c2 = __builtin_amdgcn_wmma_f32_16x16x4_f32(false, a, false, b2, (short)0, c2, false, false);
        c3 = __builtin_amdgcn_wmma_f32_16x16x4_f32(false, a, false, b3, (short)0, c3, false, false);
    }

    float* orow = xw + (size_t)(tile * 16 + 8 * half) * D + l16;
#pragma unroll
    for (int j = 0; j < 8; ++j) {
        orow[(size_t)j * D +  0] = c0[j];
        orow[(size_t)j * D + 16] = c1[j];
        orow[(size_t)j * D + 32] = c2[j];
        orow[(size_t)j * D + 48] = c3[j];
    }
}

// ---------------- edge scatter: out[dst] += dinv[s]*dinv[d] * xw[src] -------
// 16 lanes per edge, float4 gathers (xw resident in 192MB L2), 4 f32 atomics/lane.
__global__ void __launch_bounds__(256) scatter_kernel(const int* __restrict__ src,
                                                      const int* __restrict__ dst,
                                                      const float* __restrict__ dinv,
                                                      const float* __restrict__ xw,
                                                      float* __restrict__ out) {
    int t = blockIdx.x * blockDim.x + threadIdx.x;
    int e = t >> 4;
    if (e >= N_EDGES) return;
    int lane = t & 15;
    int s = src[e], d = dst[e];
    float norm = dinv[s] * dinv[d];
    float4 v = ((const float4*)(xw + (size_t)s * D))[lane];
    float* o = out + (size_t)d * D + lane * 4;
    unsafeAtomicAdd(o + 0, norm * v.x);
    unsafeAtomicAdd(o + 1, norm * v.y);
    unsafeAtomicAdd(o + 2, norm * v.z);
    unsafeAtomicAdd(o + 3, norm * v.w);
}

// ---------------- finalize: relu(agg + 2*dinv^2*xw + b) ---------------------
__global__ void finalize_kernel(const float* __restrict__ xw,
                                const float* __restrict__ dinv,
                                const float* __restrict__ b,
                                float* __restrict__ out) {
    size_t i = (size_t)blockIdx.x * blockDim.x + threadIdx.x;
    if (i >= (size_t)N_NODES * D) return;
    int node = (int)(i >> 6);
    int col  = (int)(i & 63);
    float di = dinv[node];
    float v = out[i] + 2.0f * di * di * xw[i] + b[col];
    out[i] = fmaxf(v, 0.0f);
}

extern "C" void kernel_launch(void* const* d_in, const int* in_sizes, int n_in,
                              void* d_out, int out_size, void* d_ws, size_t ws_size,
                              hipStream_t stream) {
    const float* x  = (const float*)d_in[0];
    const int*   ei = (const int*)d_in[1];      // [2, N_EDGES] flat
    const float* W  = (const float*)d_in[2];    // [64, 64] row-major
    const float* b  = (const float*)d_in[3];    // [64]
    float* out = (float*)d_out;

    float* xw   = (float*)d_ws;                        // 25.6 MB
    float* dinv = xw + (size_t)N_NODES * D;            // 0.4 MB (deg -> dinv in place)
    const int* src = ei;
    const int* dst = ei + N_EDGES;

    zero_kernel   <<<4096, 256, 0, stream>>>(out, dinv);
    degree_kernel <<<(N_EDGES + 255) / 256, 256, 0, stream>>>(dst, dinv);
    dinv_kernel   <<<(N_NODES + 255) / 256, 256, 0, stream>>>(dinv);
    gemm_wmma_kernel<<<(ROW_TILES + 7) / 8, 256, 0, stream>>>(x, W, xw);
    scatter_kernel<<<(N_EDGES * 16 + 255) / 256, 256, 0, stream>>>(src, dst, dinv, xw, out);
    finalize_kernel<<<((int)((size_t)N_NODES * D + 255) / 256), 256, 0, stream>>>(xw, dinv, b, out);
}